// MyMSELoss_35811437314683
// MI455X (gfx1250) — compile-verified
//
#include <hip/hip_runtime.h>

#define B_SIZE 4096
#define N_PTS  1024
#define DROP   102   // int(0.1 * N)

typedef float v2f __attribute__((ext_vector_type(2)));
typedef float v8f __attribute__((ext_vector_type(8)));

// ---------------------------------------------------------------------------
// Pass 1: one block (256 thr = 8 waves) per batch row.
// Computes pred via V_WMMA_F32_16X16X4_F32 (A = H 3x3 padded to 16x4,
// B = 16 homogeneous points 4x16), then w=exp(-||pred-pts2||^2) -> ws,
// plus per-block partial sums of w and the clamped dx residual.
// ---------------------------------------------------------------------------
__global__ void __launch_bounds__(256) pass1_kernel(
    const float* __restrict__ H_out, const float* __restrict__ dx,
    const float* __restrict__ pts1,  const float* __restrict__ pts2,
    float* __restrict__ w_out, float* __restrict__ partS,
    float* __restrict__ partDx)
{
    __shared__ float h[9];
    __shared__ float red_w[256];
    __shared__ float red_d[256];

    const int b   = blockIdx.x;
    const int tid = threadIdx.x;
    if (tid < 8) h[tid] = H_out[(size_t)b * 8 + tid];
    if (tid == 0) h[8] = 1.0f;
    __syncthreads();

    const int lane = tid & 31;
    const int wave = tid >> 5;

    // A-matrix (16x4 f32) per ISA layout:
    //   lanes 0-15 : VGPR0 = A[m][0], VGPR1 = A[m][1]   (m = lane)
    //   lanes16-31 : VGPR0 = A[m][2], VGPR1 = A[m][3]   (m = lane-16)
    // A rows 0..2 = rows of H, rest zero.
    float a0 = 0.0f, a1 = 0.0f;
    if      (lane == 0)  { a0 = h[0]; a1 = h[1]; }
    else if (lane == 1)  { a0 = h[3]; a1 = h[4]; }
    else if (lane == 2)  { a0 = h[6]; a1 = h[7]; }
    else if (lane == 16) { a0 = h[2]; }
    else if (lane == 17) { a0 = h[5]; }
    else if (lane == 18) { a0 = h[8]; }
    v2f Amat; Amat.x = a0; Amat.y = a1;

    const float2* dxv = (const float2*)dx;
    const float2* p1v = (const float2*)pts1;
    const float2* p2v = (const float2*)pts2;

    float wsum = 0.0f, dsum = 0.0f;

    // each wave handles 128 contiguous points, 16 per WMMA issue
    for (int i = 0; i < 8; ++i) {
        const int p = wave * 128 + i * 16;
        float sx = 0.0f, sy = 0.0f, q2x = 0.0f, q2y = 0.0f;
        const size_t idx = (size_t)b * N_PTS + (size_t)p + (size_t)lane;
        if (lane < 16) {
            float2 d  = dxv[idx];
            float2 p1 = p1v[idx];
            float2 p2 = p2v[idx];
            sx = p1.x + d.x;  sy = p1.y + d.y;
            q2x = p2.x;       q2y = p2.y;
        }
        // B-matrix (4x16 f32): lanes 0-15 rows K=0 (x), K=1 (y);
        // lanes 16-31 rows K=2 (=1), K=3 (=0).
        v2f Bmat;
        Bmat.x = (lane < 16) ? sx : 1.0f;
        Bmat.y = (lane < 16) ? sy : 0.0f;

        v8f C = {};
        // D = A x B + 0 : rows 0..2 (lanes 0-15 of D[0..2]) = tx,ty,tz per point
        v8f D = __builtin_amdgcn_wmma_f32_16x16x4_f32(
            false, Amat, false, Bmat, (short)0, C, false, false);

        if (lane < 16) {
            float inv = __builtin_amdgcn_rcpf(D[2]);   // v_rcp_f32: tz ~ 1
            float px  = D[0] * inv;
            float py  = D[1] * inv;
            float ex = px - q2x, ey = py - q2y;
            float d2 = ex * ex + ey * ey;
            float wv = __expf(-d2);
            w_out[idx] = wv;
            wsum += wv;
            float rx = fabsf(sx) - 1.0f; rx = rx > 0.0f ? rx : 0.0f;
            float ry = fabsf(sy) - 1.0f; ry = ry > 0.0f ? ry : 0.0f;
            dsum += rx + ry;
        }
    }

    red_w[tid] = wsum;
    red_d[tid] = dsum;
    __syncthreads();
    for (int s = 128; s > 0; s >>= 1) {
        if (tid < s) { red_w[tid] += red_w[tid + s]; red_d[tid] += red_d[tid + s]; }
        __syncthreads();
    }
    if (tid == 0) { partS[b] = red_w[0]; partDx[b] = red_d[0]; }
}

// ---------------------------------------------------------------------------
// Pass 2: one block per row. Radix-select the rank-(DROP-1) smallest weight
// (float bits of positive floats are monotone as u32), drop keys <= it,
// accumulate sum of w^2 * (-log w) over survivors.
// ---------------------------------------------------------------------------
__global__ void __launch_bounds__(256) pass2_kernel(
    const float* __restrict__ w_in, float* __restrict__ partMse)
{
    __shared__ float    wrow[N_PTS];
    __shared__ unsigned hist[256];
    __shared__ unsigned sh_prefix;
    __shared__ int      sh_rank;
    __shared__ float    red[256];

    const int b   = blockIdx.x;
    const int tid = threadIdx.x;
    const float* row = w_in + (size_t)b * N_PTS;

    for (int j = tid; j < N_PTS; j += 256) wrow[j] = row[j];
    if (tid == 0) { sh_prefix = 0u; sh_rank = DROP - 1; }
    __syncthreads();

    for (int shift = 24; shift >= 0; shift -= 8) {
        hist[tid] = 0u;
        __syncthreads();
        const unsigned prefix = sh_prefix;
        const unsigned hm = (shift == 24) ? 0u : (0xFFFFFFFFu << (shift + 8));
        for (int j = tid; j < N_PTS; j += 256) {
            unsigned k = __float_as_uint(wrow[j]);
            if ((k & hm) == prefix) atomicAdd(&hist[(k >> shift) & 255u], 1u);
        }
        __syncthreads();
        if (tid == 0) {
            int rank = sh_rank;
            unsigned cum = 0u;
            for (int bin = 0; bin < 256; ++bin) {
                unsigned c = hist[bin];
                if ((unsigned)rank < cum + c) {
                    sh_prefix = prefix | ((unsigned)bin << shift);
                    sh_rank   = rank - (int)cum;
                    break;
                }
                cum += c;
            }
        }
        __syncthreads();
    }

    const unsigned Kdrop = sh_prefix;  // key of the DROP-th smallest weight
    float msum = 0.0f;
    for (int j = tid; j < N_PTS; j += 256) {
        float wv   = wrow[j];
        unsigned k = __float_as_uint(wv);
        if (k > Kdrop) {
            float d2 = -__logf(wv);    // v_log_f32 path; recovers ||pred-pts2||^2
            msum += wv * wv * d2;
        }
    }
    red[tid] = msum;
    __syncthreads();
    for (int s = 128; s > 0; s >>= 1) {
        if (tid < s) red[tid] += red[tid + s];
        __syncthreads();
    }
    if (tid == 0) partMse[b] = red[0];
}

// ---------------------------------------------------------------------------
// Finalize: deterministic reduce of the three partial arrays, combine loss.
// ---------------------------------------------------------------------------
__global__ void __launch_bounds__(256) finalize_kernel(
    const float* __restrict__ partS, const float* __restrict__ partDx,
    const float* __restrict__ partMse, float* __restrict__ out)
{
    __shared__ double rs[256], rd[256], rm[256];
    const int tid = threadIdx.x;
    double s = 0.0, d = 0.0, m = 0.0;
    for (int j = tid; j < B_SIZE; j += 256) {
        s += (double)partS[j];
        d += (double)partDx[j];
        m += (double)partMse[j];
    }
    rs[tid] = s; rd[tid] = d; rm[tid] = m;
    __syncthreads();
    for (int st = 128; st > 0; st >>= 1) {
        if (tid < st) { rs[tid] += rs[tid+st]; rd[tid] += rd[tid+st]; rm[tid] += rm[tid+st]; }
        __syncthreads();
    }
    if (tid == 0) {
        double S   = rs[0];
        double mse = rm[0] / (S * S * (double)B_SIZE * (double)N_PTS * 2.0);
        out[0] = (float)(mse + rd[0] / (double)N_PTS);
    }
}

// ---------------------------------------------------------------------------
extern "C" void kernel_launch(void* const* d_in, const int* in_sizes, int n_in,
                              void* d_out, int out_size, void* d_ws, size_t ws_size,
                              hipStream_t stream)
{
    (void)in_sizes; (void)n_in; (void)out_size; (void)ws_size;
    const float* H_out = (const float*)d_in[0];
    const float* dx    = (const float*)d_in[1];
    const float* pts1  = (const float*)d_in[2];
    const float* pts2  = (const float*)d_in[3];
    float* out = (float*)d_out;

    char*  ws      = (char*)d_ws;
    float* w_arr   = (float*)ws;                                   // B*N floats (16 MB)
    float* partS   = (float*)(ws + (size_t)B_SIZE * N_PTS * sizeof(float));
    float* partDx  = partS  + B_SIZE;
    float* partMse = partDx + B_SIZE;

    pass1_kernel   <<<B_SIZE, 256, 0, stream>>>(H_out, dx, pts1, pts2,
                                                w_arr, partS, partDx);
    pass2_kernel   <<<B_SIZE, 256, 0, stream>>>(w_arr, partMse);
    finalize_kernel<<<1,      256, 0, stream>>>(partS, partDx, partMse, out);
}